// WormnetCell_44590350467249
// MI455X (gfx1250) — compile-verified
//
#include <hip/hip_runtime.h>
#include <cstdint>

// Wormnet / LTC cell for MI455X (gfx1250, wave32).
// Trans-bound elementwise+reduce; no GEMM structure -> WMMA inapplicable.
// CDNA5 paths used: TDM tensor_load_to_lds + s_wait_tensorcnt, v_tanh_f32.

#define UIN     32
#define UU      128
#define UNFOLDS 6
#define BROWS   16          // batch rows per block (== waves per block)
#define TPB     512         // 16 wave32 waves
#define BATCHN  8192

typedef unsigned int v4u __attribute__((ext_vector_type(4)));
typedef int          v8i __attribute__((ext_vector_type(8)));
typedef int          v4i __attribute__((ext_vector_type(4)));

__device__ __forceinline__ float fast_tanh(float a) {
#if __has_builtin(__builtin_amdgcn_tanhf)
  return __builtin_amdgcn_tanhf(a);              // v_tanh_f32 (gfx1250)
#else
  // tanh(a) = 1 - 2/(1 + exp(2a));  2*log2(e) = 2.8853900817779268
  float u = __builtin_amdgcn_exp2f(2.8853900817779268f * a);
  return 1.0f - 2.0f * __builtin_amdgcn_rcpf(1.0f + u);
#endif
}

// 1-D contiguous 64KB-class DMA: nelem f32 from global -> LDS byte offset.
// D# per CDNA5 ISA (count=1, data_size=4B, 1-D tile,
// unused upper dims = 1, no padding, no multicast, type=2).
__device__ __forceinline__ void tdm_load_f32(const float* gsrc,
                                             unsigned lds_byte_off,
                                             unsigned nelem) {
  unsigned long long ga = (unsigned long long)(uintptr_t)gsrc;
  v4u g0;
  g0.x = 1u;                                         // count=1, user desc
  g0.y = lds_byte_off;                               // lds_addr
  g0.z = (unsigned)(ga & 0xFFFFFFFFu);               // global_addr[31:0]
  g0.w = (unsigned)((ga >> 32) & 0x01FFFFFFu)        // global_addr[56:32]
         | 0x80000000u;                              // type=2 ("image")
  v8i g1;
  g1[0] = 0x00020000;                 // wg_mask=0, data_size=2 (4 bytes)
  g1[1] = (int)((nelem & 0xFFFFu) << 16);   // tensor_dim0 lo16 @ [63:48]
  g1[2] = (int)((nelem >> 16) & 0xFFFFu)    // tensor_dim0 hi16 @ [79:64]
          | (int)(1u << 16);                // tensor_dim1 = 1
  g1[3] = (int)((nelem & 0xFFFFu) << 16);   // tile_dim0 @ [127:112]
  g1[4] = 0;                                // tile_dim1=0, tile_dim2=0
  g1[5] = (int)nelem;                       // tensor_dim0_stride lo32
  g1[6] = (int)((nelem & 0xFFFFu) << 16);   // tensor_dim1_stride lo16
  g1[7] = 0;
  v4i g2; g2[0] = 1; g2[1] = 1; g2[2] = 0; g2[3] = 0;  // dims2/3 = 1, unused
  v4i g3; g3[0] = 0; g3[1] = (int)(1u << 16); g3[2] = 0; g3[3] = 0; // dim4=1
  v8i g4; g4[0]=0; g4[1]=0; g4[2]=0; g4[3]=0; g4[4]=0; g4[5]=0; g4[6]=0; g4[7]=0;
  // amdgpu-toolchain (clang-23) 6-arg form:
  // (uint32x4, int32x8, int32x4, int32x4, int32x8, i32 cpol)
  __builtin_amdgcn_tensor_load_to_lds(g0, g1, g2, g3, g4, 0);
}

__global__ __launch_bounds__(TPB, 1)
void wormnet_kernel(const float* __restrict__ inputs,
                    const float* __restrict__ state,
                    const float* __restrict__ smu_s,
                    const float* __restrict__ ssig_s,
                    const float* __restrict__ sW_s,
                    const float* __restrict__ serev_s,
                    const float* __restrict__ mu,
                    const float* __restrict__ sigma,
                    const float* __restrict__ W,
                    const float* __restrict__ erev,
                    const float* __restrict__ vleak,
                    const float* __restrict__ gleak,
                    const float* __restrict__ cm_t,
                    const float* __restrict__ in_w,
                    const float* __restrict__ in_b,
                    const float* __restrict__ out_w,
                    const float* __restrict__ out_b,
                    float* __restrict__ out_y,
                    float* __restrict__ out_v) {
  extern __shared__ float smem[];
  float* hsg = smem;            // [128][128]  0.5*sigma        (64KB)
  float* hsm = smem + 16384;    // [128][128]  0.5*sigma*mu     (64KB)
  float* hw  = smem + 32768;    // [128][128]  0.5*W            (64KB)
  float* hwe = smem + 49152;    // [128][128]  0.5*W*erev       (64KB)
  float* vt  = smem + 65536;    // [BROWS][128] v tile
  float* xt  = vt + BROWS * UU; // [BROWS][32]  mapped inputs
  float* Cn  = xt + BROWS * UIN;// [128] sum_i 0.5*W*erev
  float* Cd  = Cn + UU;         // [128] sum_i 0.5*W

  const int tid = threadIdx.x;
  const int l   = tid & 31;     // lane: owns columns j0..j0+3
  const int r   = tid >> 5;     // wave id == batch row within tile
  const int j0  = l << 2;

  // ---- Stage the four U*U parameter matrices into LDS via the TDM ----
  if (tid == 0) {               // EXEC-masked wave0 issues 4 TDM ops
    unsigned base = (unsigned)(uintptr_t)(void*)smem;  // low32 = LDS offset
    tdm_load_f32(sigma, base + 0u,      16384u);
    tdm_load_f32(mu,    base + 65536u,  16384u);
    tdm_load_f32(W,     base + 131072u, 16384u);
    tdm_load_f32(erev,  base + 196608u, 16384u);
    __builtin_amdgcn_s_wait_tensorcnt(0);
  }
  __syncthreads();

  // ---- Derive fused constants in place (tanh form of sigmoid) ----
  for (int k = tid; k < UU * UU; k += TPB) {
    float sg = hsg[k], m = hsm[k];
    hsm[k] = 0.5f * sg * m;     // 0.5*sigma*mu
    hsg[k] = 0.5f * sg;         // 0.5*sigma
    float w = hw[k], e = hwe[k];
    hwe[k] = 0.5f * w * e;      // 0.5*W*erev
    hw[k]  = 0.5f * w;          // 0.5*W
  }
  __syncthreads();

  // ---- Per-column constants: C_num[j]=sum_i hwe, C_den[j]=sum_i hw ----
  if (tid < 32) {
    float4 cn = {0.f, 0.f, 0.f, 0.f}, cd = {0.f, 0.f, 0.f, 0.f};
    const float4* hw4  = (const float4*)hw;
    const float4* hwe4 = (const float4*)hwe;
    for (int i = 0; i < UU; ++i) {
      float4 a = hw4[i * 32 + l];
      cd.x += a.x; cd.y += a.y; cd.z += a.z; cd.w += a.w;
      float4 b = hwe4[i * 32 + l];
      cn.x += b.x; cn.y += b.y; cn.z += b.z; cn.w += b.w;
    }
    ((float4*)Cn)[l] = cn;
    ((float4*)Cd)[l] = cd;
  }
  __syncthreads();

  // ---- Per-lane (per 4 j) leak constants ----
  float4 cm4 = *(const float4*)(cm_t + j0);
  float4 gl4 = *(const float4*)(gleak + j0);
  float4 vl4 = *(const float4*)(vleak + j0);
  float4 cn4 = ((const float4*)Cn)[l];
  float4 cd4 = ((const float4*)Cd)[l];
  float4 glvl, cgl;
  glvl.x = gl4.x * vl4.x; glvl.y = gl4.y * vl4.y;
  glvl.z = gl4.z * vl4.z; glvl.w = gl4.w * vl4.w;
  cgl.x = cm4.x + gl4.x;  cgl.y = cm4.y + gl4.y;
  cgl.z = cm4.z + gl4.z;  cgl.w = cm4.w + gl4.w;
  const float ow = out_w[0], ob = out_b[0];

  const int ntiles = BATCHN / BROWS;   // 512 tiles of 16 rows
  for (int tile = blockIdx.x; tile < ntiles; tile += gridDim.x) {
    const int row0 = tile * BROWS;

    // mapped inputs x = in*w + b   (512 elems == TPB)
    {
      int rr = tid >> 5, ii = tid & 31;
      xt[tid] = fmaf(inputs[(row0 + rr) * UIN + ii], in_w[ii], in_b[ii]);
    }
    // v tile from state
    for (int k = tid; k < BROWS * UU; k += TPB)
      vt[k] = state[row0 * UU + k];
    __syncthreads();

    // ---- sensory pass (once per tile) ----
    float4 ns = {0.f, 0.f, 0.f, 0.f}, dsn = {0.f, 0.f, 0.f, 0.f};
    for (int i = 0; i < UIN; ++i) {
      float x = xt[r * UIN + i];
      float4 m4 = *(const float4*)(smu_s  + i * UU + j0);
      float4 g4 = *(const float4*)(ssig_s + i * UU + j0);
      float4 w4 = *(const float4*)(sW_s   + i * UU + j0);
      float4 e4 = *(const float4*)(serev_s + i * UU + j0);
      float t, sg, sw;
      t = fast_tanh(0.5f * g4.x * (x - m4.x)); sg = 0.5f + 0.5f * t;
      sw = w4.x * sg; dsn.x += sw; ns.x = fmaf(sw, e4.x, ns.x);
      t = fast_tanh(0.5f * g4.y * (x - m4.y)); sg = 0.5f + 0.5f * t;
      sw = w4.y * sg; dsn.y += sw; ns.y = fmaf(sw, e4.y, ns.y);
      t = fast_tanh(0.5f * g4.z * (x - m4.z)); sg = 0.5f + 0.5f * t;
      sw = w4.z * sg; dsn.z += sw; ns.z = fmaf(sw, e4.z, ns.z);
      t = fast_tanh(0.5f * g4.w * (x - m4.w)); sg = 0.5f + 0.5f * t;
      sw = w4.w * sg; dsn.w += sw; ns.w = fmaf(sw, e4.w, ns.w);
    }
    float4 bn, bd;
    bn.x = ns.x + cn4.x + glvl.x;  bn.y = ns.y + cn4.y + glvl.y;
    bn.z = ns.z + cn4.z + glvl.z;  bn.w = ns.w + cn4.w + glvl.w;
    bd.x = dsn.x + cd4.x + cgl.x;  bd.y = dsn.y + cd4.y + cgl.y;
    bd.z = dsn.z + cd4.z + cgl.z;  bd.w = dsn.w + cd4.w + cgl.w;

    float4 vs = ((const float4*)vt)[r * 32 + l];

    // ---- 6 semi-implicit unfolds ----
    for (int st = 0; st < UNFOLDS; ++st) {
      float4 num = bn, den = bd;
      const float4* vt4 = (const float4*)vt + r * 32;
      #pragma unroll 2
      for (int i = 0; i < UU; i += 4) {
        float4 vi4 = vt4[i >> 2];                     // b128 broadcast
        float varr[4] = {vi4.x, vi4.y, vi4.z, vi4.w};
        #pragma unroll
        for (int ii = 0; ii < 4; ++ii) {
          float vi = varr[ii];
          int idx = (i + ii) * 32 + l;
          float4 hg = ((const float4*)hsg)[idx];      // conflict-free b128
          float4 hm = ((const float4*)hsm)[idx];
          float4 w4 = ((const float4*)hw)[idx];
          float4 we = ((const float4*)hwe)[idx];
          float t;
          t = fast_tanh(fmaf(hg.x, vi, -hm.x));
          num.x = fmaf(we.x, t, num.x); den.x = fmaf(w4.x, t, den.x);
          t = fast_tanh(fmaf(hg.y, vi, -hm.y));
          num.y = fmaf(we.y, t, num.y); den.y = fmaf(w4.y, t, den.y);
          t = fast_tanh(fmaf(hg.z, vi, -hm.z));
          num.z = fmaf(we.z, t, num.z); den.z = fmaf(w4.z, t, den.z);
          t = fast_tanh(fmaf(hg.w, vi, -hm.w));
          num.w = fmaf(we.w, t, num.w); den.w = fmaf(w4.w, t, den.w);
        }
      }
      float4 vn;
      vn.x = fmaf(cm4.x, vs.x, num.x) * __builtin_amdgcn_rcpf(den.x);
      vn.y = fmaf(cm4.y, vs.y, num.y) * __builtin_amdgcn_rcpf(den.y);
      vn.z = fmaf(cm4.z, vs.z, num.z) * __builtin_amdgcn_rcpf(den.z);
      vn.w = fmaf(cm4.w, vs.w, num.w) * __builtin_amdgcn_rcpf(den.w);
      __syncthreads();
      ((float4*)vt)[r * 32 + l] = vn;
      __syncthreads();
      vs = vn;
    }

    // ---- outputs: y = v[:, :1]*ow + ob, plus final v ----
    *(float4*)(out_v + (row0 + r) * UU + j0) = vs;
    if (l == 0) out_y[row0 + r] = fmaf(vs.x, ow, ob);
  }
}

extern "C" void kernel_launch(void* const* d_in, const int* in_sizes, int n_in,
                              void* d_out, int out_size, void* d_ws,
                              size_t ws_size, hipStream_t stream) {
  const float* inputs  = (const float*)d_in[0];
  const float* state   = (const float*)d_in[1];
  const float* smu_s   = (const float*)d_in[2];
  const float* ssig_s  = (const float*)d_in[3];
  const float* sW_s    = (const float*)d_in[4];
  const float* serev_s = (const float*)d_in[5];
  const float* mu      = (const float*)d_in[6];
  const float* sigma   = (const float*)d_in[7];
  const float* W       = (const float*)d_in[8];
  const float* erev    = (const float*)d_in[9];
  const float* vleak   = (const float*)d_in[10];
  const float* gleak   = (const float*)d_in[11];
  const float* cm_t    = (const float*)d_in[12];
  const float* in_w    = (const float*)d_in[13];
  const float* in_b    = (const float*)d_in[14];
  const float* out_w   = (const float*)d_in[15];
  const float* out_b   = (const float*)d_in[16];
  // d_in[17], d_in[18] are b, s scalars (256, 32) -- layout baked in.

  float* out_y = (float*)d_out;
  float* out_v = out_y + BATCHN;

  const size_t shmem = 4u * UU * UU * sizeof(float)          // 256 KB params
                     + (size_t)BROWS * UU * sizeof(float)    // v tile
                     + (size_t)BROWS * UIN * sizeof(float)   // x tile
                     + 2u * UU * sizeof(float);              // Cn, Cd
  (void)hipFuncSetAttribute(reinterpret_cast<const void*>(wormnet_kernel),
                            hipFuncAttributeMaxDynamicSharedMemorySize,
                            (int)shmem);

  dim3 grid(512), block(TPB);
  wormnet_kernel<<<grid, block, shmem, stream>>>(
      inputs, state, smu_s, ssig_s, sW_s, serev_s, mu, sigma, W, erev,
      vleak, gleak, cm_t, in_w, in_b, out_w, out_b, out_y, out_v);
}